// MAPMultilevelDense_32512902431061
// MI455X (gfx1250) — compile-verified
//
#include <hip/hip_runtime.h>
#include <hip/hip_bf16.h>

// Problem constants (match reference)
#define B_NUM 2048
#define F_DIM 512
#define U_DIM 512
#define G_NUM 64
#define XSTRIDE (F_DIM + 4)   // 516 floats: LDS bank-conflict-free (516 mod 64 == 4)

typedef float v2f __attribute__((ext_vector_type(2)));
typedef float v8f __attribute__((ext_vector_type(8)));

// ---------------- workspace layout (ints) ----------------
// [0..63]    cnt[64]
// [64..127]  cursor[64]
// [128..192] goff[65]
// [200]      numTiles
// [256..511] tiles[256]  (packed: g | (t<<8))
// [512..2559] order[2048]
// [2560..2623] (as float) S[64] = per-group ||w_g-w0||^2 + ||b_g-b0||^2

__global__ void init_kernel(int* wsI) {
    int i = threadIdx.x;
    if (i < 128) wsI[i] = 0;   // cnt + cursor
}

__global__ void count_kernel(const int* __restrict__ gid, int* __restrict__ cnt) {
    int b = blockIdx.x * blockDim.x + threadIdx.x;
    if (b < B_NUM) atomicAdd(&cnt[gid[b]], 1);
}

__global__ void plan_kernel(const int* __restrict__ cnt, int* __restrict__ goff,
                            int* __restrict__ tiles, int* __restrict__ numT) {
    if (threadIdx.x == 0 && blockIdx.x == 0) {
        int off = 0, nt = 0;
        for (int g = 0; g < G_NUM; ++g) {
            goff[g] = off;
            int c = cnt[g];
            int t16 = (c + 15) >> 4;
            for (int t = 0; t < t16; ++t) tiles[nt++] = g | (t << 8);
            off += c;
        }
        goff[G_NUM] = off;
        numT[0] = nt;
    }
}

__global__ void scatter_kernel(const int* __restrict__ gid, const int* __restrict__ goff,
                               int* __restrict__ cursor, int* __restrict__ order) {
    int b = blockIdx.x * blockDim.x + threadIdx.x;
    if (b < B_NUM) {
        int g = gid[b];
        int p = atomicAdd(&cursor[g], 1);
        order[goff[g] + p] = b;
    }
}

// per-group squared distance to prior: S[g] = ||w_g - w0||^2 + ||b_g - b0||^2
__global__ void norms_kernel(const float* __restrict__ w_mu, const float* __restrict__ b_mu,
                             const float* __restrict__ w0, const float* __restrict__ b0,
                             float* __restrict__ S) {
    int g = blockIdx.x, tid = threadIdx.x;
    const float4* wg  = (const float4*)(w_mu + (size_t)g * U_DIM * F_DIM);
    const float4* w0v = (const float4*)w0;
    float acc = 0.f;
    const int n4 = (U_DIM * F_DIM) / 4;
    for (int i = tid; i < n4; i += 256) {
        float4 a = wg[i], c = w0v[i];
        float dx = a.x - c.x, dy = a.y - c.y, dz = a.z - c.z, dw = a.w - c.w;
        acc += dx * dx + dy * dy + dz * dz + dw * dw;
    }
    const float* bg = b_mu + (size_t)g * U_DIM;
    for (int i = tid; i < U_DIM; i += 256) {
        float d = bg[i] - b0[i];
        acc += d * d;
    }
    __shared__ float red[256];
    red[tid] = acc;
    __syncthreads();
    for (int s = 128; s > 0; s >>= 1) {
        if (tid < s) red[tid] += red[tid + s];
        __syncthreads();
    }
    if (tid == 0) S[g] = red[0];
}

__global__ void finalize_kernel(const int* __restrict__ cnt, const float* __restrict__ S,
                                float* __restrict__ out_loss) {
    if (threadIdx.x == 0 && blockIdx.x == 0) {
        float t = 0.f;
        for (int g = 0; g < G_NUM; ++g) t += (float)cnt[g] * S[g];
        out_loss[0] = t;  // REG_STRENGTH = 1.0
    }
}

// Grouped GEMM: for each tile (g, t): out[16 rows of group g, u0..u0+127] via WMMA f32 16x16x4.
// Block = 256 threads = 8 wave32; wave w handles u-tile [u0 + 16w, +16).
// grid = (192 [>= worst-case tile count 188], 4 [u blocks of 128]).
__global__ void __launch_bounds__(256)
gemm_kernel(const float* __restrict__ x, const float* __restrict__ w_mu,
            const float* __restrict__ b_mu,
            const int* __restrict__ cnt, const int* __restrict__ goff,
            const int* __restrict__ tiles, const int* __restrict__ numT,
            const int* __restrict__ order, float* __restrict__ out) {
    __shared__ float sx[16 * XSTRIDE];   // 16 gathered x rows, padded stride
    __shared__ int   srow[16];

    int tileIdx = blockIdx.x;
    if (tileIdx >= numT[0]) return;
    int packed = tiles[tileIdx];
    int g = packed & 0xFF;
    int t = packed >> 8;
    int base = goff[g] + t * 16;
    int rows = cnt[g] - t * 16;
    if (rows > 16) rows = 16;

    int tid = threadIdx.x;
    if (tid < 16) srow[tid] = (tid < rows) ? order[base + tid] : -1;

    // Stage 16 x-rows into LDS (16 threads per row, float4, coalesced).
    {
        int r = tid >> 4;                 // row 0..15
        int l = tid & 15;                 // lane within row
        int gr = (r < rows) ? order[base + r] : -1;
        const float4* src = (const float4*)(x + (size_t)(gr < 0 ? 0 : gr) * F_DIM);
        #pragma unroll
        for (int j = 0; j < 8; ++j) {
            int c4 = l + j * 16;          // float4 index 0..127
            float4 v;
            if (gr >= 0) v = src[c4];
            else         v = make_float4(0.f, 0.f, 0.f, 0.f);
            *(float4*)&sx[r * XSTRIDE + c4 * 4] = v;
        }
    }
    __syncthreads();

    int wave = tid >> 5;                  // 0..7
    int lane = tid & 31;
    int n  = lane & 15;                   // N (u within tile) / M (batch row for A)
    int hl = lane >> 4;                   // half: selects K pair
    int u0 = blockIdx.y * 128 + wave * 16;

    const float* wrow = w_mu + ((size_t)g * U_DIM + (u0 + n)) * F_DIM;
    float bias = b_mu[(size_t)g * U_DIM + u0 + n];

    v8f acc;
    #pragma unroll
    for (int i = 0; i < 8; ++i) acc[i] = bias;   // C = bias (broadcast over M)

    const float* axp = &sx[n * XSTRIDE];

    // A[16x4]: lane L holds M=L&15, K = {2*hl, 2*hl+1}  (x rows from LDS)
    // B[4x16]: lane L holds N=L&15, K = {2*hl, 2*hl+1}  (w row, contiguous in f)
    #pragma unroll 8
    for (int k0 = 0; k0 < F_DIM; k0 += 4) {
        v2f a = *(const v2f*)(axp + k0 + 2 * hl);
        v2f b = *(const v2f*)(wrow + k0 + 2 * hl);
        acc = __builtin_amdgcn_wmma_f32_16x16x4_f32(
            false, a, false, b, (short)0, acc, false, false);
    }

    // D layout: VGPR r -> M = r + 8*hl, N = n
    // Hoist row-index LDS reads out of the predicated stores (2x ds_load_b128).
    int m0 = 8 * hl;
    int4 ra = *(const int4*)&srow[m0];
    int4 rb = *(const int4*)&srow[m0 + 4];
    int rid[8] = { ra.x, ra.y, ra.z, ra.w, rb.x, rb.y, rb.z, rb.w };
    float* outc = out + (u0 + n);
    #pragma unroll
    for (int r = 0; r < 8; ++r) {
        if (m0 + r < rows) outc[(size_t)rid[r] * U_DIM] = acc[r];
    }
}

extern "C" void kernel_launch(void* const* d_in, const int* in_sizes, int n_in,
                              void* d_out, int out_size, void* d_ws, size_t ws_size,
                              hipStream_t stream) {
    (void)in_sizes; (void)n_in; (void)out_size; (void)ws_size;

    const float* x    = (const float*)d_in[0];
    const int*   gid  = (const int*)d_in[1];
    const float* w_mu = (const float*)d_in[2];
    const float* b_mu = (const float*)d_in[3];
    const float* w0   = (const float*)d_in[4];
    const float* b0   = (const float*)d_in[5];

    float* out = (float*)d_out;                    // [B*U] outputs, then [1] reg loss
    int*   wsI = (int*)d_ws;
    int*   cnt    = wsI + 0;
    int*   cursor = wsI + 64;
    int*   goff   = wsI + 128;
    int*   numT   = wsI + 200;
    int*   tiles  = wsI + 256;
    int*   order  = wsI + 512;
    float* S      = (float*)(wsI + 2560);

    init_kernel<<<1, 128, 0, stream>>>(wsI);
    count_kernel<<<B_NUM / 256, 256, 0, stream>>>(gid, cnt);
    plan_kernel<<<1, 32, 0, stream>>>(cnt, goff, tiles, numT);
    scatter_kernel<<<B_NUM / 256, 256, 0, stream>>>(gid, goff, cursor, order);
    norms_kernel<<<G_NUM, 256, 0, stream>>>(w_mu, b_mu, w0, b0, S);

    dim3 grid(192, 4);
    gemm_kernel<<<grid, 256, 0, stream>>>(x, w_mu, b_mu, cnt, goff, tiles, numT, order, out);

    finalize_kernel<<<1, 32, 0, stream>>>(cnt, S, out + (size_t)B_NUM * U_DIM);
}